// VQAutoEncoder_8169027797497
// MI455X (gfx1250) — compile-verified
//
#include <hip/hip_runtime.h>
#include <math.h>

// ---------------------------------------------------------------------------
// Model dimensions (compile-time constants from the reference)
// ---------------------------------------------------------------------------
#define BB   4
#define SS   1024
#define DIN  58
#define DD   512
#define HH   8
#define HDIM 64
#define FFD  2048
#define GG   32
#define NQQ  4
#define KK   2048
#define DG   16
#define NTOK (BB * SS)        // 4096 tokens

typedef float v2f __attribute__((ext_vector_type(2)));
typedef float v8f __attribute__((ext_vector_type(8)));
typedef unsigned int u32x4 __attribute__((ext_vector_type(4)));
typedef int i32x4 __attribute__((ext_vector_type(4)));
typedef int i32x8 __attribute__((ext_vector_type(8)));

// f32 WMMA helper: D = A(16x4) * B(4x16) + C   (CDNA5 V_WMMA_F32_16X16X4_F32)
__device__ __forceinline__ v8f wmma_f32(v2f a, v2f b, v8f c) {
    return __builtin_amdgcn_wmma_f32_16x16x4_f32(false, a, false, b, (short)0, c,
                                                 false, false);
}

// ---------------------------------------------------------------------------
// Tensor Data Mover: async 2D tile load Global -> LDS (TENSORcnt tracked).
// Pads the LDS destination (pad_amount DWORDs after every pad_interval DWORDs)
// so the padded LDS strides (33 / 65 / 68) come straight out of the DMA.
// OOB reads (tile-local index beyond tensor_dim) return zero, which replaces
// the manual K/N boundary guards.
// ---------------------------------------------------------------------------
#if defined(__has_builtin)
#  if __has_builtin(__builtin_amdgcn_tensor_load_to_lds)
#    define HAVE_TDM 1
#  else
#    define HAVE_TDM 0
#  endif
#else
#  define HAVE_TDM 0
#endif

#if HAVE_TDM
__device__ __forceinline__ void tdm_load_2d(unsigned lds_off, const void* gptr,
                                            unsigned td0, unsigned td1,
                                            unsigned tile0, unsigned tile1,
                                            unsigned stride_elems,
                                            unsigned pad_interval_code,
                                            unsigned pad_amount_code) {
    unsigned long long ga = (unsigned long long)gptr;
    u32x4 g0;
    g0[0] = 1u;                                          // count=1, user mode
    g0[1] = lds_off;                                     // lds_addr (bytes)
    g0[2] = (unsigned)(ga & 0xffffffffu);                // global_addr[31:0]
    g0[3] = (unsigned)((ga >> 32) & 0x01ffffffu)         // global_addr[56:32]
            | (2u << 30);                                // type = 2 ("image")
    i32x8 g1;
    g1[0] = (int)((2u << 16)                             // data_size = 4B
                  | (1u << 20)                           // pad_enable
                  | (pad_interval_code << 22)
                  | (pad_amount_code << 25));
    g1[1] = (int)((td0 & 0xffffu) << 16);                // tensor_dim0 lo16
    g1[2] = (int)(((td0 >> 16) & 0xffffu)                // tensor_dim0 hi16
                  | ((td1 & 0xffffu) << 16));            // tensor_dim1 lo16
    g1[3] = (int)(((td1 >> 16) & 0xffffu)                // tensor_dim1 hi16
                  | (tile0 << 16));                      // tile_dim0
    g1[4] = (int)tile1;                                  // tile_dim1 (tile_dim2=0)
    g1[5] = (int)stride_elems;                           // tensor_dim0_stride lo32
    g1[6] = 0;                                           // stride hi16 | dim1_stride lo16
    g1[7] = 0;
    i32x4 z4 = {0, 0, 0, 0};
#if __clang_major__ >= 23
    i32x8 z8 = {0, 0, 0, 0, 0, 0, 0, 0};
    __builtin_amdgcn_tensor_load_to_lds(g0, g1, z4, z4, z8, 0);
#else
    __builtin_amdgcn_tensor_load_to_lds(g0, g1, z4, z4, 0);
#endif
}
#endif

// ---------------------------------------------------------------------------
// Generic tiled GEMM:  C[M,N] = act(A[M,K] @ W[K,N] + bias[N])
// Block = 256 threads (8 waves). Tile BM=64, BN=64, BK=32.
// Wave w owns the 16x32 strip at (m0=(w>>1)*16, n0=(w&1)*32): 2 accumulators,
// 16 WMMAs per BK step. Tiles staged by the TDM (padded LDS strides 33/65).
// FLAGS: bit0 = ReLU, bit1 = sin/cos positional encoding (in-proj only).
// ---------------------------------------------------------------------------
template <int FLAGS>
__global__ __launch_bounds__(256) void k_gemm(const float* __restrict__ A, int lda,
                                              const float* __restrict__ W, int ldb,
                                              const float* __restrict__ bias,
                                              float* __restrict__ C, int ldc,
                                              int K, int realN) {
    const int nBase = blockIdx.x * 64;
    const int mBase = blockIdx.y * 64;
    const int tid  = threadIdx.x;
    const int wave = tid >> 5;
    const int lane = tid & 31;
    const int half = lane >> 4;
    const int lr   = lane & 15;
    const int m0 = (wave >> 1) * 16;
    const int n0 = (wave & 1) * 32;

    __shared__ float As[64 * 33];   // 64 rows x 32 cols (+1 pad)
    __shared__ float Ws[32 * 65];   // 32 rows x 64 cols (+1 pad)

    v8f acc0 = {}, acc1 = {};

    for (int k0 = 0; k0 < K; k0 += 32) {
        __syncthreads();
#if HAVE_TDM
        if (tid < 32) {
            // A tile: 64 rows x 32 cols, zero-fill past K
            tdm_load_2d((unsigned)(size_t)(void*)As,
                        A + (size_t)mBase * lda + k0,
                        (unsigned)(K - k0), 64u, 32u, 64u, (unsigned)lda, 4u, 0u);
            // W tile: 32 rows x 64 cols, zero-fill past realN / K
            tdm_load_2d((unsigned)(size_t)(void*)Ws,
                        W + (size_t)k0 * ldb + nBase,
                        (unsigned)(realN - nBase), (unsigned)(K - k0),
                        64u, 32u, (unsigned)ldb, 5u, 0u);
            __builtin_amdgcn_s_wait_tensorcnt(0);
        }
#else
        for (int i = tid; i < 64 * 32; i += 256) {
            int rr = i >> 5, cc = i & 31;
            As[rr * 33 + cc] = (k0 + cc < K)
                ? A[(size_t)(mBase + rr) * lda + (k0 + cc)] : 0.0f;
        }
        for (int i = tid; i < 32 * 64; i += 256) {
            int rr = i >> 6, cc = i & 63;
            Ws[rr * 65 + cc] = ((k0 + rr) < K && (nBase + cc) < realN)
                ? W[(size_t)(k0 + rr) * ldb + (nBase + cc)] : 0.0f;
        }
#endif
        __syncthreads();
#pragma unroll
        for (int c = 0; c < 8; ++c) {
            v2f a, b0, b1;
            a.x = As[(m0 + lr) * 33 + 4 * c + 2 * half];
            a.y = As[(m0 + lr) * 33 + 4 * c + 2 * half + 1];
            b0.x = Ws[(4 * c + 2 * half) * 65 + n0 + lr];
            b0.y = Ws[(4 * c + 2 * half + 1) * 65 + n0 + lr];
            b1.x = Ws[(4 * c + 2 * half) * 65 + n0 + 16 + lr];
            b1.y = Ws[(4 * c + 2 * half + 1) * 65 + n0 + 16 + lr];
            acc0 = wmma_f32(a, b0, acc0);
            acc1 = wmma_f32(a, b1, acc1);
        }
    }

#pragma unroll
    for (int i = 0; i < 8; ++i) {
        int row = mBase + m0 + i + 8 * half;   // C layout: reg i -> row i (+8 hi half)
#pragma unroll
        for (int t = 0; t < 2; ++t) {
            int col = nBase + n0 + 16 * t + lr;
            float v = (t == 0) ? acc0[i] : acc1[i];
            if (col < realN) {
                v += bias[col];
                if (FLAGS & 1) v = fmaxf(v, 0.0f);
                if (FLAGS & 2) {
                    // pe[s,2i]=sin(s*div_i), pe[s,2i+1]=cos(s*div_i),
                    // div_i = exp(2i * (-ln(1e4)/D))
                    int s = row & (SS - 1);
                    int pairi = col >> 1;
                    float div = expf((float)(2 * pairi) * (-9.210340371976184f / (float)DD));
                    float ang = (float)s * div;
                    v += (col & 1) ? cosf(ang) : sinf(ang);
                }
                C[(size_t)row * ldc + col] = v;
            }
        }
    }
}

// ---------------------------------------------------------------------------
// Flash attention. qkv buffer: row-major (NTOK, 3*DD), q|k|v concatenated.
// Grid: B*H*(S/128) blocks; 8 waves, each wave owns 16 query rows.
// Shared 16x64 K and V tiles staged per key step by the TDM (stride 68 via
// pad_amount=4 DWORDs after every 64 DWORDs).
// ---------------------------------------------------------------------------
__global__ __launch_bounds__(256) void k_attn(const float* __restrict__ qkv,
                                              float* __restrict__ out) {
    const int blk = blockIdx.x;
    const int qt = blk & 7;             // S/128 = 8 query blocks
    const int h  = (blk >> 3) & (HH - 1);
    const int bb = blk >> 6;
    const int tid  = threadIdx.x;
    const int wave = tid >> 5;
    const int lane = tid & 31;
    const int half = lane >> 4;
    const int lr   = lane & 15;
    const int q0 = qt * 128 + wave * 16;

    const float* qbase = qkv + (size_t)bb * SS * (3 * DD) + h * HDIM;
    const float* kbase = qbase + DD;
    const float* vbase = qbase + 2 * DD;

    __shared__ float Ks[16 * 68];
    __shared__ float Vs[16 * 68];
    __shared__ float Ps[8 * 16 * 20];
    float* ps = Ps + wave * 16 * 20;

    // Q fragments: A-layout, 16 K-chunks of 4.
    float qa0[16], qa1[16];
#pragma unroll
    for (int c = 0; c < 16; ++c) {
        const float* qp = qbase + (size_t)(q0 + lr) * (3 * DD) + 4 * c + 2 * half;
        qa0[c] = qp[0];
        qa1[c] = qp[1];
    }

    v8f o0 = {}, o1 = {}, o2 = {}, o3 = {};
    float m[8], l[8];
#pragma unroll
    for (int i = 0; i < 8; ++i) { m[i] = -3.0e38f; l[i] = 0.0f; }

    for (int kt = 0; kt < SS / 16; ++kt) {
        __syncthreads();
#if HAVE_TDM
        if (tid < 32) {
            tdm_load_2d((unsigned)(size_t)(void*)Ks,
                        kbase + (size_t)(kt * 16) * (3 * DD),
                        64u, 16u, 64u, 16u, 3u * DD, 5u, 3u);
            tdm_load_2d((unsigned)(size_t)(void*)Vs,
                        vbase + (size_t)(kt * 16) * (3 * DD),
                        64u, 16u, 64u, 16u, 3u * DD, 5u, 3u);
            __builtin_amdgcn_s_wait_tensorcnt(0);
        }
#else
        for (int i = tid; i < 16 * 64; i += 256) {
            int rr = i >> 6, cc = i & 63;
            Ks[rr * 68 + cc] = kbase[(size_t)(kt * 16 + rr) * (3 * DD) + cc];
            Vs[rr * 68 + cc] = vbase[(size_t)(kt * 16 + rr) * (3 * DD) + cc];
        }
#endif
        __syncthreads();

        // scores S = Q @ K^T over hd=64 : 16 chained K=4 WMMAs
        v8f sc = {};
#pragma unroll
        for (int c = 0; c < 16; ++c) {
            v2f a, b;
            a.x = qa0[c]; a.y = qa1[c];
            b.x = Ks[lr * 68 + 4 * c + 2 * half];       // B[kk][n] = K[n][kk]
            b.y = Ks[lr * 68 + 4 * c + 2 * half + 1];
            sc = wmma_f32(a, b, sc);
        }

        // online softmax update (per row = per C register, within 16-lane half)
        float fac[8];
#pragma unroll
        for (int i = 0; i < 8; ++i) {
            float s = sc[i] * 0.125f;                   // 1/sqrt(64)
            float t = s;
#pragma unroll
            for (int msk = 1; msk < 16; msk <<= 1) t = fmaxf(t, __shfl_xor(t, msk, 32));
            float mn = fmaxf(m[i], t);
            float p = expf(s - mn);
            float rs = p;
#pragma unroll
            for (int msk = 1; msk < 16; msk <<= 1) rs += __shfl_xor(rs, msk, 32);
            fac[i] = expf(m[i] - mn);
            l[i] = l[i] * fac[i] + rs;
            m[i] = mn;
            ps[(i + 8 * half) * 20 + lr] = p;           // C-layout -> LDS tile
        }
#pragma unroll
        for (int i = 0; i < 8; ++i) {
            o0[i] *= fac[i]; o1[i] *= fac[i]; o2[i] *= fac[i]; o3[i] *= fac[i];
        }

        // O += P(16x16) @ V(16x64) : 4 K-chunks x 4 column tiles
#pragma unroll
        for (int c = 0; c < 4; ++c) {
            v2f a;
            a.x = ps[lr * 20 + 4 * c + 2 * half];
            a.y = ps[lr * 20 + 4 * c + 2 * half + 1];
            v2f b;
            b.x = Vs[(4 * c + 2 * half) * 68 + 0 + lr];
            b.y = Vs[(4 * c + 2 * half + 1) * 68 + 0 + lr];
            o0 = wmma_f32(a, b, o0);
            b.x = Vs[(4 * c + 2 * half) * 68 + 16 + lr];
            b.y = Vs[(4 * c + 2 * half + 1) * 68 + 16 + lr];
            o1 = wmma_f32(a, b, o1);
            b.x = Vs[(4 * c + 2 * half) * 68 + 32 + lr];
            b.y = Vs[(4 * c + 2 * half + 1) * 68 + 32 + lr];
            o2 = wmma_f32(a, b, o2);
            b.x = Vs[(4 * c + 2 * half) * 68 + 48 + lr];
            b.y = Vs[(4 * c + 2 * half + 1) * 68 + 48 + lr];
            o3 = wmma_f32(a, b, o3);
        }
    }

#pragma unroll
    for (int i = 0; i < 8; ++i) {
        float inv = 1.0f / l[i];
        int row = q0 + i + 8 * half;
        size_t base = (size_t)(bb * SS + row) * DD + h * HDIM + lr;
        out[base +  0] = o0[i] * inv;
        out[base + 16] = o1[i] * inv;
        out[base + 32] = o2[i] * inv;
        out[base + 48] = o3[i] * inv;
    }
}

// ---------------------------------------------------------------------------
// Fused residual + LayerNorm: y = LN(x + a) * g + b. One wave per token row.
// ---------------------------------------------------------------------------
__global__ __launch_bounds__(256) void k_ln_res(const float* __restrict__ x,
                                                const float* __restrict__ a,
                                                const float* __restrict__ g,
                                                const float* __restrict__ b,
                                                float* __restrict__ y) {
    const int tid = threadIdx.x;
    const int wave = tid >> 5;
    const int lane = tid & 31;
    const int row = blockIdx.x * 8 + wave;
    const float* xp = x + (size_t)row * DD;
    const float* ap = a + (size_t)row * DD;

    float v[16];
    float s = 0.0f;
#pragma unroll
    for (int j = 0; j < 16; ++j) {
        v[j] = xp[lane + 32 * j] + ap[lane + 32 * j];
        s += v[j];
    }
#pragma unroll
    for (int msk = 1; msk < 32; msk <<= 1) s += __shfl_xor(s, msk, 32);
    float mu = s * (1.0f / (float)DD);

    float vs = 0.0f;
#pragma unroll
    for (int j = 0; j < 16; ++j) { float d = v[j] - mu; vs += d * d; }
#pragma unroll
    for (int msk = 1; msk < 32; msk <<= 1) vs += __shfl_xor(vs, msk, 32);
    float inv = rsqrtf(vs * (1.0f / (float)DD) + 1e-5f);

    float* yp = y + (size_t)row * DD;
#pragma unroll
    for (int j = 0; j < 16; ++j) {
        int c = lane + 32 * j;
        yp[c] = (v[j] - mu) * inv * g[c] + b[c];
    }
}

// ---------------------------------------------------------------------------
// Codebook squared norms: norms[g,qi,k] = sum_d cb[g,qi,k,d]^2
// ---------------------------------------------------------------------------
__global__ __launch_bounds__(256) void k_cbnorm(const float* __restrict__ cb,
                                                float* __restrict__ norms) {
    int t = blockIdx.x * 256 + threadIdx.x;   // < G*NQ*K = 262144
    const float* p = cb + (size_t)t * DG;
    float s = 0.0f;
#pragma unroll
    for (int d = 0; d < DG; ++d) s += p[d] * p[d];
    norms[t] = s;
}

// ---------------------------------------------------------------------------
// One residual-VQ stage. Each wave handles (16 tokens x 1 group):
//   scores = r_tile(16x16) @ cb^T via WMMA, d2 = |cb|^2 - 2*scores,
//   per-row argmin over K=2048, then quantize: qout += cb[k*], r -= cb[k*],
//   loss += sum((cb[k*] - r_pre)^2).   Grid: (NTOK/16)*G / 8 = 1024 blocks.
// ---------------------------------------------------------------------------
__global__ __launch_bounds__(256) void k_vq(const float* __restrict__ cb,
                                            const float* __restrict__ norms,
                                            float* __restrict__ r,
                                            float* __restrict__ qout,
                                            float* __restrict__ lossacc,
                                            int qi) {
    const int tid  = threadIdx.x;
    const int wave = tid >> 5;
    const int lane = tid & 31;
    const int half = lane >> 4;
    const int lr   = lane & 15;
    const int task = blockIdx.x * 8 + wave;     // 0..8191
    const int g  = task & (GG - 1);
    const int t0 = (task >> 5) * 16;

    const float* cbq = cb    + (size_t)(g * NQQ + qi) * KK * DG;
    const float* nrm = norms + (size_t)(g * NQQ + qi) * KK;

    // residual tile A-fragments (16 tokens x DG=16)
    float a0[4], a1[4];
#pragma unroll
    for (int c = 0; c < 4; ++c) {
        const float* rp = r + (size_t)(t0 + lr) * DD + g * DG + 4 * c + 2 * half;
        a0[c] = rp[0];
        a1[c] = rp[1];
    }

    float minv[8];
    int   mini[8];
#pragma unroll
    for (int i = 0; i < 8; ++i) { minv[i] = 3.0e38f; mini[i] = 0; }

    for (int n0 = 0; n0 < KK; n0 += 16) {
        // prefetch the next 16-code tile into cache (global_prefetch_b8)
        if (n0 + 16 < KK)
            __builtin_prefetch((const void*)(cbq + (size_t)(n0 + 16 + lr) * DG), 0, 1);
        v8f sc = {};
#pragma unroll
        for (int c = 0; c < 4; ++c) {
            v2f a, b;
            a.x = a0[c]; a.y = a1[c];
            const float* cp = cbq + (size_t)(n0 + lr) * DG + 4 * c + 2 * half;
            b.x = cp[0]; b.y = cp[1];             // B[kk][n] = cb[n][kk]
            sc = wmma_f32(a, b, sc);
        }
        float nv = nrm[n0 + lr];
        int idx = n0 + lr;
#pragma unroll
        for (int i = 0; i < 8; ++i) {
            float d2 = nv - 2.0f * sc[i];
            if (d2 < minv[i]) { minv[i] = d2; mini[i] = idx; }
        }
    }

    // argmin across the 16 lanes of each half (ties -> smaller index)
#pragma unroll
    for (int i = 0; i < 8; ++i) {
#pragma unroll
        for (int msk = 1; msk < 16; msk <<= 1) {
            float ov = __shfl_xor(minv[i], msk, 32);
            int   oi = __shfl_xor(mini[i], msk, 32);
            if (ov < minv[i] || (ov == minv[i] && oi < mini[i])) {
                minv[i] = ov; mini[i] = oi;
            }
        }
    }

    // quantize: lanes 0..7 of each half handle rows lr (+8*half)
    float lsum = 0.0f;
    if (lr < 8) {
        int kstar = 0;
#pragma unroll
        for (int i = 0; i < 8; ++i) if (lr == i) kstar = mini[i];
        int row = t0 + lr + 8 * half;
        const float* q = cbq + (size_t)kstar * DG;
        float* rp = r    + (size_t)row * DD + g * DG;
        float* qp = qout + (size_t)row * DD + g * DG;
#pragma unroll
        for (int d = 0; d < DG; ++d) {
            float qd = q[d];
            float rv = rp[d];
            float df = qd - rv;
            lsum += df * df;
            qp[d] += qd;
            rp[d] = rv - qd;
        }
    }
#pragma unroll
    for (int msk = 1; msk < 32; msk <<= 1) lsum += __shfl_xor(lsum, msk, 32);
    if (lane == 0) atomicAdd(lossacc, lsum);
}

__global__ void k_loss(const float* __restrict__ lossacc, float* __restrict__ out) {
    if (threadIdx.x == 0)
        out[0] = lossacc[0] * (0.1f / (float)(NTOK * DD));   // 0.1 * sum / (B*S*G*DG), over all qi
}

// ---------------------------------------------------------------------------
// Host orchestration
// ---------------------------------------------------------------------------
static void run_block(const float* qkv_w, const float* qkv_b,
                      const float* out_w, const float* out_b,
                      const float* ln1_g, const float* ln1_b,
                      const float* ff1_w, const float* ff1_b,
                      const float* ff2_w, const float* ff2_b,
                      const float* ln2_g, const float* ln2_b,
                      float* x, float* big, float* attnb, float* t2,
                      hipStream_t stream) {
    dim3 blk(256);
    for (int i = 0; i < 6; ++i) {
        // qkv = x @ Wqkv + b
        k_gemm<0><<<dim3(3 * DD / 64, NTOK / 64), blk, 0, stream>>>(
            x, DD, qkv_w + (size_t)i * DD * 3 * DD, 3 * DD,
            qkv_b + (size_t)i * 3 * DD, big, 3 * DD, DD, 3 * DD);
        // attention
        k_attn<<<dim3(BB * HH * (SS / 128)), blk, 0, stream>>>(big, attnb);
        // out projection
        k_gemm<0><<<dim3(DD / 64, NTOK / 64), blk, 0, stream>>>(
            attnb, DD, out_w + (size_t)i * DD * DD, DD,
            out_b + (size_t)i * DD, t2, DD, DD, DD);
        // x = LN(x + attn_out)
        k_ln_res<<<dim3(NTOK / 8), blk, 0, stream>>>(
            x, t2, ln1_g + (size_t)i * DD, ln1_b + (size_t)i * DD, x);
        // ff1 = relu(x @ W1 + b1)
        k_gemm<1><<<dim3(FFD / 64, NTOK / 64), blk, 0, stream>>>(
            x, DD, ff1_w + (size_t)i * DD * FFD, FFD,
            ff1_b + (size_t)i * FFD, big, FFD, DD, FFD);
        // ff2 = ff1 @ W2 + b2
        k_gemm<0><<<dim3(DD / 64, NTOK / 64), blk, 0, stream>>>(
            big, FFD, ff2_w + (size_t)i * FFD * DD, DD,
            ff2_b + (size_t)i * DD, t2, DD, FFD, DD);
        // x = LN(x + ff2)
        k_ln_res<<<dim3(NTOK / 8), blk, 0, stream>>>(
            x, t2, ln2_g + (size_t)i * DD, ln2_b + (size_t)i * DD, x);
    }
}

extern "C" void kernel_launch(void* const* d_in, const int* in_sizes, int n_in,
                              void* d_out, int out_size, void* d_ws, size_t ws_size,
                              hipStream_t stream) {
    const float* blend = (const float*)d_in[0];
    const float* w_in  = (const float*)d_in[1];
    const float* b_in  = (const float*)d_in[2];
    const float* e[12];
    const float* d[12];
    for (int i = 0; i < 12; ++i) e[i] = (const float*)d_in[3 + i];
    for (int i = 0; i < 12; ++i) d[i] = (const float*)d_in[15 + i];
    const float* cbk   = (const float*)d_in[27];
    const float* w_out = (const float*)d_in[28];
    const float* b_out = (const float*)d_in[29];
    float* out = (float*)d_out;

    // workspace layout (floats)
    float* ws      = (float*)d_ws;
    float* x       = ws;                          // NTOK*DD
    float* big     = x + (size_t)NTOK * DD;       // NTOK*FFD (shared qkv/ff1)
    float* attnb   = big + (size_t)NTOK * FFD;    // NTOK*DD
    float* t2      = attnb + (size_t)NTOK * DD;   // NTOK*DD
    float* rbuf    = t2 + (size_t)NTOK * DD;      // NTOK*DD (VQ residual)
    float* qbuf    = rbuf + (size_t)NTOK * DD;    // NTOK*DD (VQ output / dec x)
    float* norms   = qbuf + (size_t)NTOK * DD;    // G*NQ*K
    float* lossacc = norms + (size_t)GG * NQQ * KK;

    dim3 blk(256);

    // 1. input projection + positional encoding
    k_gemm<2><<<dim3(DD / 64, NTOK / 64), blk, 0, stream>>>(
        blend, DIN, w_in, DD, b_in, x, DD, DIN, DD);

    // 2. encoder
    run_block(e[0], e[1], e[2], e[3], e[4], e[5], e[6], e[7], e[8], e[9], e[10], e[11],
              x, big, attnb, t2, stream);

    // 3. grouped residual VQ
    hipMemcpyAsync(rbuf, x, (size_t)NTOK * DD * sizeof(float),
                   hipMemcpyDeviceToDevice, stream);
    hipMemsetAsync(qbuf, 0, (size_t)NTOK * DD * sizeof(float), stream);
    hipMemsetAsync(lossacc, 0, sizeof(float), stream);
    k_cbnorm<<<dim3(GG * NQQ * KK / 256), blk, 0, stream>>>(cbk, norms);
    for (int qi = 0; qi < NQQ; ++qi)
        k_vq<<<dim3((NTOK / 16) * GG / 8), blk, 0, stream>>>(
            cbk, norms, rbuf, qbuf, lossacc, qi);

    // 4. decoder (operates on quantized output)
    run_block(d[0], d[1], d[2], d[3], d[4], d[5], d[6], d[7], d[8], d[9], d[10], d[11],
              qbuf, big, attnb, t2, stream);

    // 5. output projection (N padded 58 -> 64, store-guarded), straight to d_out
    k_gemm<0><<<dim3(1, NTOK / 64), blk, 0, stream>>>(
        qbuf, DD, w_out, DIN, b_out, out, DIN, DD, DIN);

    // 6. loss scalar appended after decoded output
    k_loss<<<dim3(1), blk, 0, stream>>>(lossacc, out + (size_t)NTOK * DIN);
}